// SelfAttentionHead_64338610095063
// MI455X (gfx1250) — compile-verified
//
#include <hip/hip_runtime.h>
#include <math.h>

typedef __attribute__((ext_vector_type(2))) float v2f;
typedef __attribute__((ext_vector_type(8))) float v8f;

__device__ __forceinline__ v8f wmma_f32(v2f a, v2f b, v8f c) {
    // D = A(16x4,f32) * B(4x16,f32) + C(16x16,f32), full fp32 precision
    return __builtin_amdgcn_wmma_f32_16x16x4_f32(
        /*neg_a=*/false, a, /*neg_b=*/false, b,
        /*c_mod=*/(short)0, c, /*reuse_a=*/false, /*reuse_b=*/false);
}

// CDNA5 async global->LDS copy (ASYNCcnt tracked, bypasses VGPRs).
// VDST vgpr = wave-relative LDS byte address, VADDR = 64-bit global address.
__device__ __forceinline__ void async_copy_b128(const void* gptr, void* lptr) {
    const uint64_t ga = (uint64_t)(uintptr_t)gptr;
    const uint32_t la = (uint32_t)(uintptr_t)lptr;  // low 32 bits = LDS offset
    asm volatile("global_load_async_to_lds_b128 %0, %1, off"
                 :: "v"(la), "v"(ga) : "memory");
}
__device__ __forceinline__ void wait_async0() {
    asm volatile("s_wait_asynccnt 0x0" ::: "memory");
}

// ---------------------------------------------------------------------------
// GEMM: C[b][m][n] = sum_k A(m,k) * W[n*K + k] + bias[n]
//   AMODE 0: A stored [b][K][M]  (x tensor: [b][c][h*w])
//   AMODE 1: A stored [b][M][K]  (attn tensor: [b][s][c])
// Tile: 64(M) x 64(N) per 256-thread block, KC=16 per stage.
// Staging via async b128 copies, double-buffered against the WMMA stream.
// ---------------------------------------------------------------------------
template <int AMODE>
__global__ __launch_bounds__(256) void gemm_wmma_f32(
    const float* __restrict__ A, const float* __restrict__ W,
    const float* __restrict__ bias, float* __restrict__ C,
    int M, int N, int K) {
    // Row strides chosen for 16B alignment (b128) + conflict-free frag reads:
    //  Ak0: stride 88 (88%64=24 -> half-waves 48 banks apart)
    //  As1/Bs: stride 20 (20*l16 mod 64 all distinct)
    __shared__ float Ak0[2][16][88];  // AMODE 0: k-major (async copy order)
    __shared__ float As1[2][64][20];  // AMODE 1: m-major
    __shared__ float Bs[2][64][20];

    const int b = blockIdx.z;
    const float* Ab = A + (size_t)b * M * K;
    float* Cb = C + (size_t)b * M * N;
    const int m0 = blockIdx.x * 64;
    const int n0 = blockIdx.y * 64;

    const int tid  = threadIdx.x;
    const int wave = tid >> 5;
    const int lane = tid & 31;
    const int half = lane >> 4;
    const int l16  = lane & 15;
    const int msub  = wave & 3;         // 0..3 : which 16-row slice
    const int nsub0 = (wave >> 2) * 2;  // 0 or 2: first of two 16-col slices

    // per-thread staging coordinates
    const int a0_kk = tid >> 4, a0_mg = tid & 15;  // AMODE 0: 16 rows x 16 quads
    const int a1_m  = tid >> 2, a1_kg = tid & 3;   // AMODE 1: 64 rows x 4 quads
    const int b_n   = tid >> 2, b_kg  = tid & 3;

    auto stage = [&](int k0, int buf) {
        if (AMODE == 0) {
            async_copy_b128(&Ab[(size_t)(k0 + a0_kk) * M + m0 + a0_mg * 4],
                            &Ak0[buf][a0_kk][a0_mg * 4]);
        } else {
            async_copy_b128(&Ab[(size_t)(m0 + a1_m) * K + k0 + a1_kg * 4],
                            &As1[buf][a1_m][a1_kg * 4]);
        }
        async_copy_b128(&W[(size_t)(n0 + b_n) * K + k0 + b_kg * 4],
                        &Bs[buf][b_n][b_kg * 4]);
    };

    v8f acc0 = {};
    v8f acc1 = {};

    stage(0, 0);
    wait_async0();
    __syncthreads();

    for (int k0 = 0; k0 < K; k0 += 16) {
        const int cur = (k0 >> 4) & 1;
        // kick off next chunk's DMA while we compute this one
        if (k0 + 16 < K) stage(k0 + 16, cur ^ 1);

#pragma unroll
        for (int ks = 0; ks < 16; ks += 4) {
            v2f a;
            if (AMODE == 0) {
                a.x = Ak0[cur][ks + 2 * half][msub * 16 + l16];
                a.y = Ak0[cur][ks + 2 * half + 1][msub * 16 + l16];
            } else {
                a.x = As1[cur][msub * 16 + l16][ks + 2 * half];
                a.y = As1[cur][msub * 16 + l16][ks + 2 * half + 1];
            }
            v2f b0;
            b0.x = Bs[cur][nsub0 * 16 + l16][ks + 2 * half];
            b0.y = Bs[cur][nsub0 * 16 + l16][ks + 2 * half + 1];
            v2f b1;
            b1.x = Bs[cur][(nsub0 + 1) * 16 + l16][ks + 2 * half];
            b1.y = Bs[cur][(nsub0 + 1) * 16 + l16][ks + 2 * half + 1];
            acc0 = wmma_f32(a, b0, acc0);
            acc1 = wmma_f32(a, b1, acc1);
        }

        wait_async0();     // own async ops done (next buffer filled for me)
        __syncthreads();   // everyone done reading cur + writing next
    }

    // epilogue: C layout = vgpr r -> row (r + 8*half), col = lane&15
#pragma unroll
    for (int r = 0; r < 8; ++r) {
        const int m  = m0 + msub * 16 + r + 8 * half;
        const int na = n0 + nsub0 * 16 + l16;
        const int nb = na + 16;
        Cb[(size_t)m * N + na] = acc0[r] + bias[na];
        Cb[(size_t)m * N + nb] = acc1[r] + bias[nb];
    }
}

// ---------------------------------------------------------------------------
// Attention: per (batch, head, 32-row q-block):
//   S = (Q_blk @ K_h^T) * 1/sqrt(hd)  -> LDS [32][1024]
//   P = softmax_rows(S)  (1/rowsum deferred to epilogue)
//   O = P @ V_h          -> attn buffer [b][s][c]
// ---------------------------------------------------------------------------
__global__ __launch_bounds__(256) void attn_wmma_f32(
    const float* __restrict__ q, const float* __restrict__ k,
    const float* __restrict__ v, float* __restrict__ o) {
    constexpr int SEQ = 1024, C = 512, HD = 64;
    constexpr int QB = 32;               // q rows per workgroup
    constexpr float SCALE = 0.125f;      // 1/sqrt(64)

    __shared__ float Sl[QB][SEQ + 4];    // stride % 64 == 4 -> conflict-free frags
    __shared__ float red[256];
    __shared__ float rowmax32[QB];
    __shared__ float rowsum32[QB];

    const int qb    = blockIdx.x * QB;
    const int head  = blockIdx.y;
    const int batch = blockIdx.z;

    const float* Qb = q + (size_t)batch * SEQ * C + head * HD;
    const float* Kb = k + (size_t)batch * SEQ * C + head * HD;
    const float* Vb = v + (size_t)batch * SEQ * C + head * HD;
    float*       Ob = o + (size_t)batch * SEQ * C + head * HD;

    const int tid  = threadIdx.x;
    const int wave = tid >> 5;
    const int lane = tid & 31;
    const int half = lane >> 4;
    const int l16  = lane & 15;

    // ---- S = Q @ K^T ----
    // wave w: q-subtile qs = w&1 (16 rows), n-tile range [(w>>1)*16, +16)
    {
        const int qs  = wave & 1;
        const int nt0 = (wave >> 1) * 16;

        // preload this wave's Q fragments (16 k-steps of K=4)
        v2f aq[16];
#pragma unroll
        for (int ks = 0; ks < 16; ++ks) {
            aq[ks].x = Qb[(size_t)(qb + qs * 16 + l16) * C + ks * 4 + 2 * half];
            aq[ks].y = Qb[(size_t)(qb + qs * 16 + l16) * C + ks * 4 + 2 * half + 1];
        }

        for (int nt = nt0; nt < nt0 + 16; ++nt) {
            const int t = nt * 16 + l16;  // key row, B col = lane
            // pull the next tile's K rows toward the WGP early
            if (nt + 1 < nt0 + 16)
                __builtin_prefetch(&Kb[(size_t)(t + 16) * C], 0, 1);
            v8f cs = {};
#pragma unroll
            for (int ks = 0; ks < 16; ++ks) {
                v2f bk;
                bk.x = Kb[(size_t)t * C + ks * 4 + 2 * half];
                bk.y = Kb[(size_t)t * C + ks * 4 + 2 * half + 1];
                cs = wmma_f32(aq[ks], bk, cs);
            }
#pragma unroll
            for (int r = 0; r < 8; ++r)
                Sl[qs * 16 + r + 8 * half][nt * 16 + l16] = cs[r] * SCALE;
        }
    }
    __syncthreads();

    // ---- row softmax over 1024 cols (8 threads per row, 32 rows) ----
    const int row  = tid >> 3;
    const int col0 = tid & 7;
    float mx = -3.4e38f;
    for (int c0 = col0; c0 < SEQ; c0 += 8) mx = fmaxf(mx, Sl[row][c0]);
    red[tid] = mx;
    __syncthreads();
    if (col0 == 0) {
        float m2 = red[tid];
#pragma unroll
        for (int j = 1; j < 8; ++j) m2 = fmaxf(m2, red[tid + j]);
        rowmax32[row] = m2;
    }
    __syncthreads();
    const float rm = rowmax32[row];
    float sm = 0.0f;
    for (int c0 = col0; c0 < SEQ; c0 += 8) {
        const float e = __expf(Sl[row][c0] - rm);
        Sl[row][c0] = e;
        sm += e;
    }
    red[tid] = sm;
    __syncthreads();
    if (col0 == 0) {
        float s2 = red[tid];
#pragma unroll
        for (int j = 1; j < 8; ++j) s2 += red[tid + j];
        rowsum32[row] = s2;
    }
    __syncthreads();

    // ---- O = P @ V : 2 M-subtiles x 4 N-subtiles across 8 waves,
    //      each wave runs the full K=1024 reduction ----
    {
        const int msub = wave & 1;       // which 16-row slice of the 32 q rows
        const int nsub = wave >> 1;      // which 16-col slice of hd=64
        v8f co = {};
#pragma unroll 4
        for (int kk = 0; kk < SEQ; kk += 4) {
            v2f ap;
            ap.x = Sl[msub * 16 + l16][kk + 2 * half];
            ap.y = Sl[msub * 16 + l16][kk + 2 * half + 1];
            v2f bv;
            bv.x = Vb[(size_t)(kk + 2 * half) * C + nsub * 16 + l16];
            bv.y = Vb[(size_t)(kk + 2 * half + 1) * C + nsub * 16 + l16];
            co = wmma_f32(ap, bv, co);
        }
#pragma unroll
        for (int r = 0; r < 8; ++r) {
            const int m = msub * 16 + r + 8 * half;
            const float val = co[r] / rowsum32[m];
            Ob[(size_t)(qb + m) * C + nsub * 16 + l16] = val;
        }
    }
}

// ---------------------------------------------------------------------------
extern "C" void kernel_launch(void* const* d_in, const int* in_sizes, int n_in,
                              void* d_out, int out_size, void* d_ws, size_t ws_size,
                              hipStream_t stream) {
    constexpr int B = 8, SEQ = 1024, C = 512;
    const float* x  = (const float*)d_in[0];
    const float* Wq = (const float*)d_in[1];
    const float* bq = (const float*)d_in[2];
    const float* Wk = (const float*)d_in[3];
    const float* bk = (const float*)d_in[4];
    const float* Wv = (const float*)d_in[5];
    const float* bv = (const float*)d_in[6];
    const float* Wo = (const float*)d_in[7];
    const float* bo = (const float*)d_in[8];
    float* out = (float*)d_out;

    const size_t per = (size_t)B * SEQ * C;  // 4M floats = 16 MB
    float* qbuf = (float*)d_ws;
    float* kbuf = qbuf + per;
    float* vbuf = kbuf + per;
    float* abuf = vbuf + per;

    dim3 blk(256);
    dim3 gemm_grid(SEQ / 64, C / 64, B);  // 16 x 8 x 8

    // Q/K/V projections: A = x stored [b][c][s] -> AMODE 0
    gemm_wmma_f32<0><<<gemm_grid, blk, 0, stream>>>(x, Wq, bq, qbuf, SEQ, C, C);
    gemm_wmma_f32<0><<<gemm_grid, blk, 0, stream>>>(x, Wk, bk, kbuf, SEQ, C, C);
    gemm_wmma_f32<0><<<gemm_grid, blk, 0, stream>>>(x, Wv, bv, vbuf, SEQ, C, C);

    // attention per (32-row q-block, head, batch)
    attn_wmma_f32<<<dim3(SEQ / 32, 8, B), blk, 0, stream>>>(qbuf, kbuf, vbuf, abuf);

    // output projection: A = attn stored [b][s][c] -> AMODE 1
    gemm_wmma_f32<1><<<gemm_grid, blk, 0, stream>>>(abuf, Wo, bo, out, SEQ, C, C);
}